// AnchorDetector_13486197310044
// MI455X (gfx1250) — compile-verified
//
#include <hip/hip_runtime.h>
#include <math.h>

typedef float v2f  __attribute__((ext_vector_type(2)));
typedef float v8f  __attribute__((ext_vector_type(8)));
typedef unsigned int u32x4 __attribute__((ext_vector_type(4)));
typedef int   i32x8 __attribute__((ext_vector_type(8)));
typedef int   i32x4 __attribute__((ext_vector_type(4)));

#define BATCH   4
#define T_LEN   8192
#define DM      1024
#define ROW_PAD 68   // 64 dwords data + 4 dwords TDM padding

__device__ __forceinline__ unsigned rfl(unsigned x) {
  return (unsigned)__builtin_amdgcn_readfirstlane((int)x);
}

// TDM: load a 17-row x 64-col f32 tile (row stride 1024 elems in memory) to LDS,
// with LDS padding of 4 dwords every 64 dwords (row stride 68 dwords in LDS).
__device__ __forceinline__ void tdm_load_tile(unsigned lds_off, unsigned long long ga) {
  u32x4 g0;
  g0[0] = 1u;                                   // count=1, user descriptor
  g0[1] = rfl(lds_off);                         // lds_addr (bytes)
  g0[2] = rfl((unsigned)ga);                    // global_addr[31:0]
  g0[3] = rfl((unsigned)(ga >> 32)) | 0x80000000u; // global_addr[56:32] | type=2
  i32x8 g1;
  g1[0] = (int)((2u << 16)      // data_size = 4B
              | (1u << 20)      // pad_enable
              | (5u << 22)      // pad_interval: 64 dwords
              | (3u << 25));    // pad_amount: 4 dwords
  g1[1] = (int)(64u << 16);     // tensor_dim0 = 64 (low16 at [63:48])
  g1[2] = (int)(17u << 16);     // dim0 hi16 = 0 | tensor_dim1 = 17 (low16)
  g1[3] = (int)(64u << 16);     // dim1 hi16 = 0 | tile_dim0 = 64
  g1[4] = 17;                   // tile_dim1 = 17 | tile_dim2 = 0
  g1[5] = 1024;                 // tensor_dim0_stride low32 (elements)
  g1[6] = 0;
  g1[7] = 0;
  i32x4 gz4 = {0, 0, 0, 0};
  i32x8 gz8 = {0, 0, 0, 0, 0, 0, 0, 0};
  __builtin_amdgcn_tensor_load_to_lds(g0, g1, gz4, gz4, gz8, 0);
}

// ---------------------------------------------------------------------------
// Stage 1: stream hidden, compute prior_raw[b,t] = hidden.W  and
//          run_sq[b,t] = ||hidden[t]-hidden[t-1]||^2 via WMMA f32 16x16x4.
// One wave per 16-row tile; TDM double-buffered tiles of 17 rows x 64 cols.
// ---------------------------------------------------------------------------
__global__ __launch_bounds__(128) void anchor_stage1(
    const float* __restrict__ hidden, const float* __restrict__ W,
    float* __restrict__ prior_raw, float* __restrict__ run_sq) {
  __shared__ float ldsW[DM];
  __shared__ float tiles[4][2][17 * ROW_PAD];

  const int tid = threadIdx.x;
  for (int i = tid; i < DM; i += 128) ldsW[i] = W[i];
  __syncthreads();

  const int wave = tid >> 5;
  const int lane = tid & 31;
  const int tileId = blockIdx.x * 4 + wave;   // 0..2047
  const int b  = tileId >> 9;                 // / (T_LEN/16)
  const int t0 = (tileId & 511) << 4;
  const int rowStart = (t0 > 0) ? (t0 - 1) : 0;

  const unsigned long long gbase =
      (unsigned long long)(const void*)(hidden + ((size_t)b * T_LEN + rowStart) * DM);
  const unsigned ldsBase0 = (unsigned)(size_t)(void*)&tiles[wave][0][0];
  const unsigned ldsBase1 = (unsigned)(size_t)(void*)&tiles[wave][1][0];

  const int i16  = lane & 15;
  const int half = lane >> 4;
  const int curSlot  = (t0 == 0) ? i16 : (i16 + 1);
  const int prevSlot = (t0 == 0) ? ((i16 > 0) ? (i16 - 1) : 0) : i16;

  v8f cp = {0.f, 0.f, 0.f, 0.f, 0.f, 0.f, 0.f, 0.f};
  v8f cr = {0.f, 0.f, 0.f, 0.f, 0.f, 0.f, 0.f, 0.f};

  auto compute_stage = [&](const float* buf, const float* wrow) {
#pragma unroll 4
    for (int k4 = 0; k4 < 16; ++k4) {
      const int kk = k4 * 4 + half * 2;
      v2f acur  = *(const v2f*)&buf[curSlot  * ROW_PAD + kk];
      v2f aprev = *(const v2f*)&buf[prevSlot * ROW_PAD + kk];
      v2f ad = acur - aprev;
      v2f wf = *(const v2f*)&wrow[kk];  // broadcast: B[k][n] = W[k] for all n
      cp = __builtin_amdgcn_wmma_f32_16x16x4_f32(false, acur, false, wf,
                                                 (short)0, cp, false, false);
      // B-fragment of delta^T equals the A-fragment of delta (layout identity):
      cr = __builtin_amdgcn_wmma_f32_16x16x4_f32(false, ad, false, ad,
                                                 (short)0, cr, false, false);
    }
  };

  tdm_load_tile(ldsBase0, gbase);
  for (int s = 0; s < 15; ++s) {
    tdm_load_tile((s & 1) ? ldsBase0 : ldsBase1,
                  gbase + (unsigned long long)(s + 1) * 256ull);
    __builtin_amdgcn_s_wait_tensorcnt(1);
    compute_stage((s & 1) ? &tiles[wave][1][0] : &tiles[wave][0][0], &ldsW[s * 64]);
  }
  __builtin_amdgcn_s_wait_tensorcnt(0);
  compute_stage(&tiles[wave][1][0], &ldsW[15 * 64]);

  // Diagonal extraction: lane r (r<8) holds (M=r,N=r); lane r+24 holds (M=r+8,N=r+8).
  const size_t outBase = (size_t)b * T_LEN + t0;
#pragma unroll
  for (int r = 0; r < 8; ++r) {
    if (lane == r || lane == r + 24) {
      const int m = r + 8 * half;
      prior_raw[outBase + m] = cp[r];
      run_sq[outBase + m]    = cr[r];
    }
  }
}

// ---------------------------------------------------------------------------
// Stage 2: per-batch standardize + sigmoids + span bounds.
// ---------------------------------------------------------------------------
__global__ __launch_bounds__(256) void anchor_stage2(
    const float* __restrict__ prior_raw, const float* __restrict__ run_sq,
    const float* __restrict__ bias, float* __restrict__ out) {
  const int b = blockIdx.x;
  const int tid = threadIdx.x;
  __shared__ float red[4][256];

  const float bb = bias[0];
  float sp = 0.f, sp2 = 0.f, sr = 0.f, sr2 = 0.f;
  for (int t = tid; t < T_LEN; t += 256) {
    float p = prior_raw[b * T_LEN + t] + bb;
    float r = sqrtf(run_sq[b * T_LEN + t]) * (1.0f / 32.0f);
    sp += p; sp2 += p * p; sr += r; sr2 += r * r;
  }
  red[0][tid] = sp; red[1][tid] = sp2; red[2][tid] = sr; red[3][tid] = sr2;
  __syncthreads();
  for (int s = 128; s > 0; s >>= 1) {
    if (tid < s) {
      red[0][tid] += red[0][tid + s];
      red[1][tid] += red[1][tid + s];
      red[2][tid] += red[2][tid + s];
      red[3][tid] += red[3][tid + s];
    }
    __syncthreads();
  }
  const float inv = 1.0f / (float)T_LEN;
  const float mp = red[0][0] * inv;
  const float sdp = fmaxf(sqrtf(fmaxf(red[1][0] * inv - mp * mp, 0.f)), 1e-6f);
  const float mr = red[2][0] * inv;
  const float sdr = fmaxf(sqrtf(fmaxf(red[3][0] * inv - mr * mr, 0.f)), 1e-6f);

  float* scores  = out;
  float* weights = out + (size_t)BATCH * T_LEN;
  float* priorS  = out + 2 * (size_t)BATCH * T_LEN;
  float* runS    = out + 3 * (size_t)BATCH * T_LEN;
  int*   span    = (int*)(out + 4 * (size_t)BATCH * T_LEN);

  for (int t = tid; t < T_LEN; t += 256) {
    float p = prior_raw[b * T_LEN + t] + bb;
    float r = sqrtf(run_sq[b * T_LEN + t]) * (1.0f / 32.0f);
    float pl = (p - mp) / sdp;
    float rl = (r - mr) / sdr;
    float cl = 0.5f * pl + 0.5f * rl;
    const size_t o = (size_t)b * T_LEN + t;
    scores[o]  = 1.f / (1.f + __expf(-cl));
    weights[o] = cl;
    priorS[o]  = 1.f / (1.f + __expf(-pl));
    runS[o]    = 1.f / (1.f + __expf(-rl));
    span[o * 2 + 0] = (t > 0) ? (t - 1) : 0;
    span[o * 2 + 1] = t;
  }
}

// ---------------------------------------------------------------------------
// Stage 3: peak detection + top-64 (bitonic sort of packed score|idx keys).
// ---------------------------------------------------------------------------
__global__ __launch_bounds__(1024) void anchor_stage3(float* __restrict__ out) {
  const int b = blockIdx.x;
  const int tid = threadIdx.x;
  __shared__ unsigned keys[T_LEN];
  __shared__ int sidx[64];
  __shared__ int sval[64];

  const float* scores = out;
  for (int i = tid; i < T_LEN; i += 1024) {
    float s = scores[b * T_LEN + i];
    float l = scores[b * T_LEN + ((i > 0) ? (i - 1) : 0)];
    float r = scores[b * T_LEN + ((i < T_LEN - 1) ? (i + 1) : (T_LEN - 1))];
    unsigned q = 0u;
    if ((s >= l) && (s >= r) && (s >= 0.6f)) {
      float sc = fminf(fmaxf(s, 0.f), 1.f);
      unsigned qs = (unsigned)(sc * 262143.f);
      if (qs > 262143u) qs = 262143u;
      q = (qs << 13) | (unsigned)(T_LEN - 1 - i);  // tie-break: lower idx wins
    }
    keys[i] = q;
  }
  __syncthreads();

  for (unsigned k = 2; k <= (unsigned)T_LEN; k <<= 1) {
    for (unsigned j = k >> 1; j > 0; j >>= 1) {
      for (unsigned i = tid; i < (unsigned)T_LEN; i += 1024) {
        unsigned ixj = i ^ j;
        if (ixj > i) {
          unsigned a = keys[i], c = keys[ixj];
          bool descend = ((i & k) == 0);
          if (descend ? (a < c) : (a > c)) { keys[i] = c; keys[ixj] = a; }
        }
      }
      __syncthreads();
    }
  }

  int*   cand = (int*)(out + 4 * (size_t)BATCH * T_LEN) + 2 * (size_t)BATCH * T_LEN;
  float* mask = (float*)(cand + BATCH * 64);
  if (tid < 64) {
    unsigned key = keys[tid];
    int v = (key != 0u) ? 1 : 0;
    int idx = v ? (T_LEN - 1 - (int)(key & 0x1FFFu)) : T_LEN;
    sidx[tid] = idx;
    sval[tid] = v;
    mask[b * 64 + tid] = (float)v;
    cand[b * 64 + tid] = T_LEN;  // default pad
  }
  __syncthreads();
  if (tid < 64 && sval[tid]) {
    int my = sidx[tid];
    int rank = 0;
    for (int j = 0; j < 64; ++j) rank += (sval[j] && (sidx[j] < my)) ? 1 : 0;
    cand[b * 64 + rank] = my;
  }
}

// ---------------------------------------------------------------------------
extern "C" void kernel_launch(void* const* d_in, const int* in_sizes, int n_in,
                              void* d_out, int out_size, void* d_ws, size_t ws_size,
                              hipStream_t stream) {
  const float* hidden = (const float*)d_in[0];
  const float* W      = (const float*)d_in[1];
  const float* bias   = (const float*)d_in[2];
  float* out = (float*)d_out;

  float* prior_raw = (float*)d_ws;
  float* run_sq    = prior_raw + (size_t)BATCH * T_LEN;

  // 2048 wave-tiles (16 rows each), 4 waves per 128-thread block.
  anchor_stage1<<<512, 128, 0, stream>>>(hidden, W, prior_raw, run_sq);
  anchor_stage2<<<BATCH, 256, 0, stream>>>(prior_raw, run_sq, bias, out);
  anchor_stage3<<<BATCH, 1024, 0, stream>>>(out);
}